// GraphSAGE_37890201485515
// MI455X (gfx1250) — compile-verified
//
#include <hip/hip_runtime.h>

#define NN 50000
#define NE 800000
#define NFEAT 128
#define NHID 256
#define NCLASS 64

typedef float v2f __attribute__((ext_vector_type(2)));
typedef float v8f __attribute__((ext_vector_type(8)));

// ---------------- utility: zero a region (float4 granularity) ----------------
__global__ void zero_f4(float4* __restrict__ p, long n4) {
    long i = (long)blockIdx.x * blockDim.x + threadIdx.x;
    long stride = (long)gridDim.x * blockDim.x;
    for (; i < n4; i += stride) p[i] = make_float4(0.f, 0.f, 0.f, 0.f);
}

// ---------------- degree count (float, for mean divide) ----------------------
__global__ void count_edges(const int* __restrict__ dst, float* __restrict__ cnt) {
    int e = blockIdx.x * blockDim.x + threadIdx.x;
    if (e < NE) atomicAdd(&cnt[dst[e]], 1.0f);
}

// ---------------- edge gather + scatter-add (segment sum) -------------------
// lane-fast over feature chunks -> coalesced float4 gathers of full rows,
// hardware GLOBAL_ATOMIC_ADD_F32 scatters.
template <int D>
__global__ void scatter_add(const float* __restrict__ feat,
                            const int* __restrict__ src,
                            const int* __restrict__ dst,
                            float* __restrict__ sum) {
    const int chunks = D / 4;
    long idx = (long)blockIdx.x * blockDim.x + threadIdx.x;
    if (idx >= (long)NE * chunks) return;
    int e = (int)(idx / chunks);
    int c = (int)(idx % chunks);
    int s = src[e];
    int d = dst[e];
    float4 v = ((const float4*)(feat + (size_t)s * D))[c];
    float* o = sum + (size_t)d * D + (size_t)c * 4;
    atomicAdd(o + 0, v.x);
    atomicAdd(o + 1, v.y);
    atomicAdd(o + 2, v.z);
    atomicAdd(o + 3, v.w);
}

// ---------------- fused dual GEMM: out = mean@Wn + root@Wr + b --------------
// One block per 16-node row tile. A tiles staged in LDS (mean computed on the
// fly from sum/cnt). NOUT/16 waves, each owning one 16x16 C tile, accumulate
// with V_WMMA_F32_16X16X4_F32 (exact fp32 matrix pipe).
template <int K, int NOUT, bool RELU>
__global__ __launch_bounds__(32 * (NOUT / 16)) void sage_gemm(
    const float* __restrict__ sumbuf, const float* __restrict__ cnt,
    const float* __restrict__ root, const float* __restrict__ Wn,
    const float* __restrict__ Wr, const float* __restrict__ bias,
    float* __restrict__ out) {
    constexpr int NT = NOUT / 16;   // waves per block
    constexpr int KP = K + 4;       // padded LDS stride (bank-conflict free)
    __shared__ float lAm[16 * KP];  // mean tile
    __shared__ float lAr[16 * KP];  // root tile

    const int mTile = blockIdx.x;
    const int tid = threadIdx.x;
    const int lane = tid & 31;
    const int wave = tid >> 5;

    // ---- cooperative staging of the 16 x K A-tiles ----
    const int vecPerRow = K / 4;
    const int totalVec = 16 * vecPerRow;
    for (int v = tid; v < totalVec; v += 32 * NT) {
        int row = v / vecPerRow;
        int c4 = v % vecPerRow;
        int node = mTile * 16 + row;
        float inv = 1.0f / fmaxf(cnt[node], 1.0f);
        float4 s = ((const float4*)(sumbuf + (size_t)node * K))[c4];
        float4 m = make_float4(s.x * inv, s.y * inv, s.z * inv, s.w * inv);
        ((float4*)(lAm + row * KP))[c4] = m;
        ((float4*)(lAr + row * KP))[c4] =
            ((const float4*)(root + (size_t)node * K))[c4];
    }
    __syncthreads();

    // ---- per-wave 16x16 tile ----
    const int nBase = wave * 16;
    const int mA = lane & 15;          // A row for this lane
    const int kh = (lane >> 4) * 2;    // K sub-offset (0 or 2) per ISA layout
    const int nB = lane & 15;          // C/D/B column for this lane

    // bias preloaded into accumulator: C[m][n] starts at bias[n] (same column
    // for all 8 accumulator rows of a lane)
    v8f acc;
    float bv = bias[nBase + nB];
#pragma unroll
    for (int r = 0; r < 8; ++r) acc[r] = bv;

    // mean @ Wn
#pragma unroll 4
    for (int k0 = 0; k0 < K; k0 += 4) {
        v2f a, b;
        a[0] = lAm[mA * KP + k0 + kh + 0];
        a[1] = lAm[mA * KP + k0 + kh + 1];
        b[0] = Wn[(size_t)(k0 + kh + 0) * NOUT + nBase + nB];
        b[1] = Wn[(size_t)(k0 + kh + 1) * NOUT + nBase + nB];
        acc = __builtin_amdgcn_wmma_f32_16x16x4_f32(false, a, false, b,
                                                    (short)0, acc, false, false);
    }
    // root @ Wr
#pragma unroll 4
    for (int k0 = 0; k0 < K; k0 += 4) {
        v2f a, b;
        a[0] = lAr[mA * KP + k0 + kh + 0];
        a[1] = lAr[mA * KP + k0 + kh + 1];
        b[0] = Wr[(size_t)(k0 + kh + 0) * NOUT + nBase + nB];
        b[1] = Wr[(size_t)(k0 + kh + 1) * NOUT + nBase + nB];
        acc = __builtin_amdgcn_wmma_f32_16x16x4_f32(false, a, false, b,
                                                    (short)0, acc, false, false);
    }

    // ---- store: acc[r] = C[r + 8*(lane/16)][nB] ----
    const int rowOff = (lane >> 4) * 8;
#pragma unroll
    for (int r = 0; r < 8; ++r) {
        int node = mTile * 16 + rowOff + r;
        float v = acc[r];
        if (RELU) v = fmaxf(v, 0.0f);
        out[(size_t)node * NOUT + nBase + nB] = v;
    }
}

// ---------------- workspace layout (floats) ----------------------------------
// cnt   : [0, 50176)                       (padded)
// sum1  : [50176, 50176 + 50000*128)
// sum2  : [sum1_end, +50000*256)
// h     : [sum2_end, +50000*256)
#define WS_CNT 0L
#define WS_SUM1 50176L
#define WS_SUM2 (WS_SUM1 + (long)NN * NFEAT)
#define WS_H (WS_SUM2 + (long)NN * NHID)
#define WS_ZERO_FLOATS WS_H  // cnt + sum1 + sum2 must be zeroed each call

extern "C" void kernel_launch(void* const* d_in, const int* in_sizes, int n_in,
                              void* d_out, int out_size, void* d_ws,
                              size_t ws_size, hipStream_t stream) {
    const float* x = (const float*)d_in[0];
    const int* edge = (const int*)d_in[1];
    const float* W1n = (const float*)d_in[2];
    const float* W1r = (const float*)d_in[3];
    const float* b1 = (const float*)d_in[4];
    const float* W2n = (const float*)d_in[5];
    const float* W2r = (const float*)d_in[6];
    const float* b2 = (const float*)d_in[7];
    float* out = (float*)d_out;

    const int* src = edge;
    const int* dst = edge + NE;

    float* ws = (float*)d_ws;
    float* cnt = ws + WS_CNT;
    float* sum1 = ws + WS_SUM1;
    float* sum2 = ws + WS_SUM2;
    float* h = ws + WS_H;

    // 1) zero accumulation buffers (every call: deterministic)
    zero_f4<<<2048, 256, 0, stream>>>((float4*)ws, WS_ZERO_FLOATS / 4);

    // 2) degree counts
    count_edges<<<(NE + 255) / 256, 256, 0, stream>>>(dst, cnt);

    // 3) layer-1 neighbor sum: sum1[d] += x[s]
    {
        long total = (long)NE * (NFEAT / 4);
        scatter_add<NFEAT>
            <<<(int)((total + 255) / 256), 256, 0, stream>>>(x, src, dst, sum1);
    }

    // 4) h = relu(mean1 @ W1n + x @ W1r + b1)   [50000 x 256]
    sage_gemm<NFEAT, NHID, true>
        <<<NN / 16, 32 * (NHID / 16), 0, stream>>>(sum1, cnt, x, W1n, W1r, b1, h);

    // 5) layer-2 neighbor sum: sum2[d] += h[s]
    {
        long total = (long)NE * (NHID / 4);
        scatter_add<NHID>
            <<<(int)((total + 255) / 256), 256, 0, stream>>>(h, src, dst, sum2);
    }

    // 6) out = mean2 @ W2n + h @ W2r + b2   [50000 x 64]
    sage_gemm<NHID, NCLASS, false>
        <<<NN / 16, 32 * (NCLASS / 16), 0, stream>>>(sum2, cnt, h, W2n, W2r, b2,
                                                     out);
}